// GCNEncoder_69157563400469
// MI455X (gfx1250) — compile-verified
//
#include <hip/hip_runtime.h>

// GCN 2-layer forward for MI455X (gfx1250, wave32).
// Memory/atomic-bound workload: GEMMs done exactly in fp32 via
// V_WMMA_F32_16X16X4_F32 (B-operand staged in LDS), edge gather/scatter via
// coalesced lane-per-column fp32 global atomics that resolve in the 192MB L2.

typedef __attribute__((ext_vector_type(2))) float v2f;
typedef __attribute__((ext_vector_type(8))) float v8f;

#define IN_DIM 128
#define H1_DIM 16
#define H2_DIM 8

// ---------------------------------------------------------------------------
// Degree / normalization
// ---------------------------------------------------------------------------
__global__ void gcn_init_deg(float* __restrict__ deg, int n) {
    int i = blockIdx.x * blockDim.x + threadIdx.x;
    if (i < n) deg[i] = 1.0f;  // self-loop contributes 1
}

__global__ void gcn_count_deg(const int* __restrict__ dst, float* __restrict__ deg, int e) {
    int i = blockIdx.x * blockDim.x + threadIdx.x;
    if (i < e) atomicAdd(&deg[dst[i]], 1.0f);
}

__global__ void gcn_rsqrt_deg(float* __restrict__ deg, int n) {
    int i = blockIdx.x * blockDim.x + threadIdx.x;
    if (i < n) {
        float d = deg[i];
        deg[i] = (d > 0.0f) ? rsqrtf(d) : 0.0f;  // deg >= 1 always; guard matches ref
    }
}

// ---------------------------------------------------------------------------
// Layer 1 GEMM: hw1 = x @ W1   (N x 128) @ (128 x 16) in exact fp32 WMMA.
// 8 waves/block, one 16-row tile per wave; W1 staged once in LDS (8 KB).
// Fuses the self-loop + bias init: agg1 = dis^2 * hw1 + b1.
//
// A 16x4 f32: lanes 0-15 row=lane  {V0=K0,V1=K1}; lanes 16-31 row=lane-16 {V0=K2,V1=K3}
// B 4x16 f32: lanes 0-15 col=lane  {V0=K0,V1=K1}; lanes 16-31 col=lane-16 {V0=K2,V1=K3}
// D 16x16 f32: VGPR r, lane<16 -> (M=r, N=lane); lane>=16 -> (M=r+8, N=lane-16)
// ---------------------------------------------------------------------------
__global__ __launch_bounds__(256) void gcn_gemm1_wmma(
    const float* __restrict__ x, const float* __restrict__ W1,
    const float* __restrict__ dis, const float* __restrict__ b1,
    float* __restrict__ hw1, float* __restrict__ agg1, int n) {
    __shared__ float W1s[IN_DIM * H1_DIM];  // 8 KB
    const int tid = threadIdx.x;
    #pragma unroll
    for (int j = 0; j < (IN_DIM * H1_DIM) / 256; ++j)
        W1s[tid + j * 256] = W1[tid + j * 256];
    __syncthreads();

    const int wave = tid >> 5;
    const int lane = tid & 31;              // EXEC all ones within a live wave
    const int half = lane >> 4;             // 0 | 1
    const int l    = lane & 15;
    const int row0 = (blockIdx.x * 8 + wave) * 16;
    if (row0 >= n) return;                  // wave-uniform exit

    int arow = row0 + l;
    if (arow >= n) arow = n - 1;            // tail clamp (keeps EXEC full)

    v8f c = {};
    #pragma unroll 4
    for (int k0 = 0; k0 < IN_DIM; k0 += 4) {
        const int ka = k0 + half * 2;
        const v2f a = *(const v2f*)&x[(size_t)arow * IN_DIM + ka];  // b64 load
        v2f b;
        b.x = W1s[ka * H1_DIM + l];
        b.y = W1s[(ka + 1) * H1_DIM + l];
        c = __builtin_amdgcn_wmma_f32_16x16x4_f32(
                false, a, false, b, (short)0, c, false, false);
    }

    const float bias = b1[l];
    if (row0 + 16 <= n) {                   // full tile: straight-line epilogue
        #pragma unroll
        for (int r = 0; r < 8; ++r) {
            const int row = row0 + r + half * 8;
            const float v  = c[r];
            const float dv = dis[row];
            hw1 [(size_t)row * H1_DIM + l] = v;
            agg1[(size_t)row * H1_DIM + l] = dv * dv * v + bias;
        }
    } else {
        for (int r = 0; r < 8; ++r) {
            const int row = row0 + r + half * 8;
            if (row < n) {
                const float v  = c[r];
                const float dv = dis[row];
                hw1 [(size_t)row * H1_DIM + l] = v;
                agg1[(size_t)row * H1_DIM + l] = dv * dv * v + bias;
            }
        }
    }
}

// ---------------------------------------------------------------------------
// Edge scatter, layer 1: agg1[dst] += dis[src]*dis[dst] * hw1[src]
// One lane per (edge, column): 16 lanes cooperate on one edge, so the gather
// is a coalesced 64B half-wave load and each atomic instruction covers two
// contiguous 64B row segments. agg1/hw1 are L2-resident -> on-chip atomics.
// ---------------------------------------------------------------------------
__global__ __launch_bounds__(256) void gcn_scatter1(
    const int* __restrict__ src, const int* __restrict__ dst,
    const float* __restrict__ dis, const float* __restrict__ hw1,
    float* __restrict__ agg1, int e) {
    const int i = blockIdx.x * blockDim.x + threadIdx.x;
    if (i >= e * H1_DIM) return;
    const int eidx = i >> 4;
    const int c    = i & 15;

    // Index-lookahead prefetch of a future gathered row (global_prefetch_b8);
    // one lane per edge-group issues it.
    if (c == 0) {
        const int la = eidx + 512;
        if (la < e) __builtin_prefetch(&hw1[(size_t)src[la] * H1_DIM], 0, 0);
    }

    const int s = src[eidx];
    const int d = dst[eidx];
    const float w = dis[s] * dis[d];
    atomicAdd(&agg1[(size_t)d * H1_DIM + c], w * hw1[(size_t)s * H1_DIM + c]);
}

// ---------------------------------------------------------------------------
// Layer 2 GEMM: hw2 = relu(agg1) @ W2   (N x 16) @ (16 x 8); W2 zero-padded
// to 16x16 in LDS. ReLU fused into the A-operand load. Fuses the output init:
// out = dis^2 * hw2 + b2 (fully overwrites poisoned d_out).
// ---------------------------------------------------------------------------
__global__ __launch_bounds__(256) void gcn_gemm2_wmma(
    const float* __restrict__ h1, const float* __restrict__ W2,
    const float* __restrict__ dis, const float* __restrict__ b2,
    float* __restrict__ hw2, float* __restrict__ out, int n) {
    __shared__ float W2s[H1_DIM * 16];      // 16x16 padded, 1 KB
    const int tid = threadIdx.x;
    {
        const int k = tid >> 4, col = tid & 15;
        W2s[tid] = (col < H2_DIM) ? W2[k * H2_DIM + col] : 0.0f;
    }
    __syncthreads();

    const int wave = tid >> 5;
    const int lane = tid & 31;
    const int half = lane >> 4;
    const int l    = lane & 15;
    const int row0 = (blockIdx.x * 8 + wave) * 16;
    if (row0 >= n) return;

    int arow = row0 + l;
    if (arow >= n) arow = n - 1;

    v8f c = {};
    #pragma unroll
    for (int k0 = 0; k0 < H1_DIM; k0 += 4) {
        const int ka = k0 + half * 2;
        v2f a = *(const v2f*)&h1[(size_t)arow * H1_DIM + ka];
        a.x = fmaxf(a.x, 0.0f);             // ReLU on load
        a.y = fmaxf(a.y, 0.0f);
        v2f b;
        b.x = W2s[ka * 16 + l];
        b.y = W2s[(ka + 1) * 16 + l];
        c = __builtin_amdgcn_wmma_f32_16x16x4_f32(
                false, a, false, b, (short)0, c, false, false);
    }

    if (l < H2_DIM) {
        const float bias = b2[l];
        if (row0 + 16 <= n) {
            #pragma unroll
            for (int r = 0; r < 8; ++r) {
                const int row = row0 + r + half * 8;
                const float v  = c[r];
                const float dv = dis[row];
                hw2[(size_t)row * H2_DIM + l] = v;
                out[(size_t)row * H2_DIM + l] = dv * dv * v + bias;
            }
        } else {
            for (int r = 0; r < 8; ++r) {
                const int row = row0 + r + half * 8;
                if (row < n) {
                    const float v  = c[r];
                    const float dv = dis[row];
                    hw2[(size_t)row * H2_DIM + l] = v;
                    out[(size_t)row * H2_DIM + l] = dv * dv * v + bias;
                }
            }
        }
    }
}

// ---------------------------------------------------------------------------
// Edge scatter, layer 2: out[dst] += dis[src]*dis[dst] * hw2[src]  (8 cols)
// One lane per (edge, column): 8 lanes per edge, 4 edges per wave.
// ---------------------------------------------------------------------------
__global__ __launch_bounds__(256) void gcn_scatter2(
    const int* __restrict__ src, const int* __restrict__ dst,
    const float* __restrict__ dis, const float* __restrict__ hw2,
    float* __restrict__ out, int e) {
    const int i = blockIdx.x * blockDim.x + threadIdx.x;
    if (i >= e * H2_DIM) return;
    const int eidx = i >> 3;
    const int c    = i & 7;

    if (c == 0) {
        const int la = eidx + 512;
        if (la < e) __builtin_prefetch(&hw2[(size_t)src[la] * H2_DIM], 0, 0);
    }

    const int s = src[eidx];
    const int d = dst[eidx];
    const float w = dis[s] * dis[d];
    atomicAdd(&out[(size_t)d * H2_DIM + c], w * hw2[(size_t)s * H2_DIM + c]);
}

// ---------------------------------------------------------------------------
extern "C" void kernel_launch(void* const* d_in, const int* in_sizes, int n_in,
                              void* d_out, int out_size, void* d_ws, size_t ws_size,
                              hipStream_t stream) {
    const float* x    = (const float*)d_in[0];
    const int*   edge = (const int*)  d_in[1];
    const float* W1   = (const float*)d_in[2];
    const float* b1   = (const float*)d_in[3];
    const float* W2   = (const float*)d_in[4];
    const float* b2   = (const float*)d_in[5];
    float* out = (float*)d_out;

    const int n = in_sizes[0] / IN_DIM;   // 100000
    const int e = in_sizes[1] / 2;        // 6400000
    const int* src = edge;                // edge_index row 0
    const int* dst = edge + e;            // edge_index row 1

    // Workspace (floats): dis[n] | hw1[n*16] | agg1[n*16] | hw2[n*8]  (~16.4 MB)
    float* ws   = (float*)d_ws;
    float* dis  = ws;
    float* hw1  = dis + n;
    float* agg1 = hw1 + (size_t)n * H1_DIM;
    float* hw2  = agg1 + (size_t)n * H1_DIM;

    const int B = 256;
    const int gn     = (n + B - 1) / B;
    const int ge     = (e + B - 1) / B;
    const int tiles  = (n + 15) / 16;             // 6250
    const int gtiles = (tiles + 7) / 8;           // 8 waves (tiles) per block
    const int gs1    = (e * H1_DIM + B - 1) / B;  // lane per (edge, col), layer 1
    const int gs2    = (e * H2_DIM + B - 1) / B;  // lane per (edge, col), layer 2

    gcn_init_deg <<<gn, B, 0, stream>>>(dis, n);
    gcn_count_deg<<<ge, B, 0, stream>>>(dst, dis, e);
    gcn_rsqrt_deg<<<gn, B, 0, stream>>>(dis, n);

    gcn_gemm1_wmma<<<gtiles, B, 0, stream>>>(x, W1, dis, b1, hw1, agg1, n);
    gcn_scatter1  <<<gs1, B, 0, stream>>>(src, dst, dis, hw1, agg1, e);

    gcn_gemm2_wmma<<<gtiles, B, 0, stream>>>(agg1, W2, dis, b2, hw2, out, n);
    gcn_scatter2  <<<gs2, B, 0, stream>>>(src, dst, dis, hw2, out, e);
}